// GraphConvolution_30322469110220
// MI455X (gfx1250) — compile-verified
//
#include <hip/hip_runtime.h>
#include <hip/hip_bf16.h>

typedef __attribute__((ext_vector_type(2))) float v2f;
typedef __attribute__((ext_vector_type(4))) float v4f;
typedef __attribute__((ext_vector_type(8))) float v8f;

#define D_DIM 256

// ---------------------------------------------------------------------------
// Kernel 1: support = X @ W  (fp32 GEMM via V_WMMA_F32_16X16X4_F32)
// One wave computes a 16-row x 256-col output strip. (unchanged from round 1;
// emits clause'd loads + 16 independent WMMA accumulation chains)
// ---------------------------------------------------------------------------
__global__ __launch_bounds__(256) void gemm_xw(const float* __restrict__ X,
                                               const float* __restrict__ W,
                                               float* __restrict__ S,
                                               int nRows) {
    const int lane  = threadIdx.x & 31;
    const int wave  = threadIdx.x >> 5;
    const int strip = blockIdx.x * 8 + wave;           // 16-row strip index
    if (strip * 16 >= nRows) return;                   // wave-uniform: EXEC stays all-1

    const int row16 = strip * 16;
    const int mrow  = lane & 15;                       // M for A, N for B
    const int khalf = (lane >> 4) * 2;                 // 0 for lanes 0-15, 2 for 16-31

    v8f acc[16];
#pragma unroll
    for (int i = 0; i < 16; ++i) acc[i] = (v8f)0.0f;

    const float* __restrict__ Arow = X + (size_t)(row16 + mrow) * D_DIM;

    for (int k0 = 0; k0 < D_DIM; k0 += 4) {
        v2f a;
        a.x = Arow[k0 + khalf];
        a.y = Arow[k0 + khalf + 1];
#pragma unroll
        for (int nt = 0; nt < 16; ++nt) {
            const float* __restrict__ Bp =
                W + (size_t)(k0 + khalf) * D_DIM + nt * 16 + mrow;
            v2f b;
            b.x = Bp[0];
            b.y = Bp[D_DIM];
            acc[nt] = __builtin_amdgcn_wmma_f32_16x16x4_f32(
                false, a, false, b, (short)0, acc[nt], false, false);
        }
    }

    const int srow = row16 + (lane >> 4) * 8;
    const int scol = lane & 15;
#pragma unroll
    for (int nt = 0; nt < 16; ++nt) {
#pragma unroll
        for (int v = 0; v < 8; ++v) {
            S[(size_t)(srow + v) * D_DIM + nt * 16 + scol] = acc[nt][v];
        }
    }
}

// ---------------------------------------------------------------------------
// CSR construction: counts -> exclusive scan -> fill (atomic cursor on rowPtr)
// ---------------------------------------------------------------------------
__global__ __launch_bounds__(256) void zero_counts(int* __restrict__ counts, int n) {
    int i = blockIdx.x * blockDim.x + threadIdx.x;
    if (i < n) counts[i] = 0;
}

__global__ __launch_bounds__(256) void count_rows(const int* __restrict__ row,
                                                  int* __restrict__ counts, int nEdges) {
    int e = blockIdx.x * blockDim.x + threadIdx.x;
    if (e < nEdges) atomicAdd(&counts[row[e]], 1);
}

// Per-block exclusive scan (1024 elems/block) + block totals.
__global__ __launch_bounds__(1024) void scan_block(const int* __restrict__ in,
                                                   int* __restrict__ outExcl,
                                                   int* __restrict__ blockSums, int n) {
    __shared__ int temp[1024];
    const int tid = threadIdx.x;
    const int i = blockIdx.x * 1024 + tid;
    const int x = (i < n) ? in[i] : 0;
    temp[tid] = x;
    for (int off = 1; off < 1024; off <<= 1) {
        __syncthreads();
        const int t = (tid >= off) ? temp[tid - off] : 0;
        __syncthreads();
        temp[tid] += t;
    }
    __syncthreads();
    if (i < n) outExcl[i] = temp[tid] - x;             // exclusive
    if (tid == 1023) blockSums[blockIdx.x] = temp[tid]; // block total
}

// Exclusive scan of the block totals (single block; supports up to 1024 blocks).
__global__ __launch_bounds__(1024) void scan_tops(int* __restrict__ blockSums, int nb) {
    __shared__ int temp[1024];
    const int tid = threadIdx.x;
    const int x = (tid < nb) ? blockSums[tid] : 0;
    temp[tid] = x;
    for (int off = 1; off < 1024; off <<= 1) {
        __syncthreads();
        const int t = (tid >= off) ? temp[tid - off] : 0;
        __syncthreads();
        temp[tid] += t;
    }
    __syncthreads();
    if (tid < nb) blockSums[tid] = temp[tid] - x;      // exclusive
}

__global__ __launch_bounds__(1024) void scan_addback(int* __restrict__ outExcl,
                                                     const int* __restrict__ blockSums,
                                                     int n) {
    const int i = blockIdx.x * 1024 + threadIdx.x;
    if (i < n) outExcl[i] += blockSums[blockIdx.x];
}

// p = rowPtr[row[e]]++ ; edgeIdx[p] = e.  Afterwards rowPtr[i] == row-i END.
__global__ __launch_bounds__(256) void fill_csr(const int* __restrict__ row,
                                                int* __restrict__ rowPtr,
                                                int* __restrict__ edgeIdx, int nEdges) {
    int e = blockIdx.x * blockDim.x + threadIdx.x;
    if (e < nEdges) {
        int p = atomicAdd(&rowPtr[row[e]], 1);
        edgeIdx[p] = e;
    }
}

// ---------------------------------------------------------------------------
// Gather-reduce: one wave per node; lane holds 8 output floats as 2 x float4.
// Per edge: 2 global_load_b128 per lane (coalesced 1KB row), FMA into regs.
// Zero output atomics. 2-edge unroll overlaps dependent-load chains.
// ---------------------------------------------------------------------------
__global__ __launch_bounds__(256) void spmm_gather(const float* __restrict__ S,
                                                   const float* __restrict__ vals,
                                                   const int* __restrict__ col,
                                                   const int* __restrict__ rowEnd,
                                                   const int* __restrict__ counts,
                                                   const int* __restrict__ edgeIdx,
                                                   float* __restrict__ out, int nNodes) {
    const int lane = threadIdx.x & 31;
    const int wave = threadIdx.x >> 5;
    const int node = blockIdx.x * 8 + wave;
    if (node >= nNodes) return;

    const int end   = rowEnd[node];
    const int cnt   = counts[node];
    const int start = end - cnt;

    const v4f* __restrict__ S4 = (const v4f*)S;   // 64 float4 per row
    v4f a0 = (v4f)0.0f, a1 = (v4f)0.0f;

    int k = start;
    for (; k + 1 < end; k += 2) {
        const int e0 = edgeIdx[k];
        const int e1 = edgeIdx[k + 1];
        const float v0 = vals[e0];
        const float v1 = vals[e1];
        const size_t b0 = (size_t)col[e0] * 64;
        const size_t b1 = (size_t)col[e1] * 64;
        const v4f s00 = S4[b0 + lane];
        const v4f s01 = S4[b0 + 32 + lane];
        const v4f s10 = S4[b1 + lane];
        const v4f s11 = S4[b1 + 32 + lane];
        a0 += v0 * s00;
        a1 += v0 * s01;
        a0 += v1 * s10;
        a1 += v1 * s11;
    }
    if (k < end) {
        const int e = edgeIdx[k];
        const float v = vals[e];
        const size_t b = (size_t)col[e] * 64;
        a0 += v * S4[b + lane];
        a1 += v * S4[b + 32 + lane];
    }

    v4f* out4 = (v4f*)out;
    const size_t o = (size_t)node * 64;
    out4[o + lane] = a0;
    out4[o + 32 + lane] = a1;
}

// ---------------------------------------------------------------------------
extern "C" void kernel_launch(void* const* d_in, const int* in_sizes, int n_in,
                              void* d_out, int out_size, void* d_ws, size_t ws_size,
                              hipStream_t stream) {
    const float* X    = (const float*)d_in[0];   // [N, 256]
    const float* W    = (const float*)d_in[1];   // [256, 256]
    const float* vals = (const float*)d_in[2];   // [E]
    const int*   row  = (const int*)d_in[3];     // [E]
    const int*   col  = (const int*)d_in[4];     // [E]
    float*       out  = (float*)d_out;           // [N, 256]

    const int nNodes = in_sizes[0] / D_DIM;
    const int nEdges = in_sizes[2];

    // Workspace carve-up.
    float* S        = (float*)d_ws;                       // N*256 floats
    int*   counts   = (int*)(S + (size_t)nNodes * D_DIM); // N
    int*   rowPtr   = counts + nNodes;                    // N
    int*   edgeIdx  = rowPtr + nNodes;                    // E
    int*   blkSums  = edgeIdx + nEdges;                   // <=1024

    // 1) support = X @ W  (WMMA)
    const int strips = (nNodes + 15) / 16;
    gemm_xw<<<(strips + 7) / 8, 256, 0, stream>>>(X, W, S, nNodes);

    // 2) CSR build: counts, exclusive scan -> rowPtr, fill (rowPtr -> row ends)
    zero_counts<<<(nNodes + 255) / 256, 256, 0, stream>>>(counts, nNodes);
    count_rows<<<(nEdges + 255) / 256, 256, 0, stream>>>(row, counts, nEdges);

    const int nb = (nNodes + 1023) / 1024;                // 98 for N=100000
    scan_block<<<nb, 1024, 0, stream>>>(counts, rowPtr, blkSums, nNodes);
    scan_tops<<<1, 1024, 0, stream>>>(blkSums, nb);
    scan_addback<<<nb, 1024, 0, stream>>>(rowPtr, blkSums, nNodes);

    fill_csr<<<(nEdges + 255) / 256, 256, 0, stream>>>(row, rowPtr, edgeIdx, nEdges);

    // 3) out[node] = sum_{e in row(node)} vals[e] * support[col[e]]
    spmm_gather<<<(nNodes + 7) / 8, 256, 0, stream>>>(S, vals, col, rowPtr, counts,
                                                      edgeIdx, out, nNodes);
}